// GCN_is_male_1425929142658
// MI455X (gfx1250) — compile-verified
//
#include <hip/hip_runtime.h>
#include <math.h>

#define N_NODES 20000
#define N_EDGES 320000
#define N_GRAPH 64

typedef __attribute__((ext_vector_type(2))) float v2f;
typedef __attribute__((ext_vector_type(8))) float v8f;

// ---------------- utility fills / padding ----------------
__global__ void fill_f32(float* __restrict__ p, int n, float v) {
    int i = blockIdx.x * blockDim.x + threadIdx.x;
    if (i < n) p[i] = v;
}

// dst is rowsPad x ldd, zero-filled outside the rows x cols source region (plain layout)
__global__ void pad_copy(const float* __restrict__ src, float* __restrict__ dst,
                         int rows, int cols, int ldd) {
    const int c = blockIdx.x * blockDim.x + threadIdx.x;
    const int r = blockIdx.y;
    if (c < ldd)
        dst[(long)r * ldd + c] = (r < rows && c < cols) ? src[(long)r * cols + c] : 0.0f;
}

// Pack weights for WMMA: row pairs interleaved per column, zero-padded.
// dst[((r/2)*ldd + c)*2 + (r%2)] = src[r, c]; launched over rowsP x ldd (rowsP%4==0, ldd%64==0)
__global__ void pad_pack_W(const float* __restrict__ src, float* __restrict__ dst,
                           int rows, int cols, int ldd) {
    const int c = blockIdx.x * blockDim.x + threadIdx.x;
    const int r = blockIdx.y;
    if (c < ldd) {
        const float v = (r < rows && c < cols) ? src[(long)r * cols + c] : 0.0f;
        dst[(((long)(r >> 1) * ldd + c) << 1) | (r & 1)] = v;
    }
}

// deg count via float atomics (accumulated into dinv buffer)
__global__ void count_deg(const long long* __restrict__ dst, float* __restrict__ deg, int e) {
    int i = blockIdx.x * blockDim.x + threadIdx.x;
    if (i < e) atomicAdd(&deg[(int)dst[i]], 1.0f);
}

__global__ void make_dinv(float* __restrict__ d, int n) {
    int i = blockIdx.x * blockDim.x + threadIdx.x;
    if (i < n) d[i] = rsqrtf(d[i] + 1.0f);
}

// ---------------- FP32 WMMA GEMM: C[M,Nc] = A[M,K] * Bp[K,Nc](packed) ----------------
// Preconditions: M%32==0, K%4==0 (K even), Nc%64==0. Bp packed by pad_pack_W.
// One wave computes a 32(M) x 64(N) tile with eight V_WMMA_F32_16X16X4_F32 accumulators:
// per K-step, 2 A b64-loads + 4 B b64-loads feed 8 WMMAs (B amortized over 2 row tiles).
// blockDim = (32, 4): 4 waves -> 128 rows per block. Branch-free hot loop.
__global__ void gemm_wmma_f32(const float* __restrict__ A, const float* __restrict__ Bp,
                              float* __restrict__ C, int M, int K, int Nc) {
    const int lane = threadIdx.x;                        // 0..31 (wave32)
    const int row0 = (blockIdx.x * blockDim.y + threadIdx.y) << 5;
    const int col0 = (int)blockIdx.y << 6;
    if (row0 >= M) return;                               // wave-uniform exit

    const int half = lane >> 4;                          // 0: K0/K1 pair, 1: K2/K3 pair
    const int l15  = lane & 15;
    const int ak   = half << 1;

    unsigned aoff0 = (unsigned)(row0 + l15) * (unsigned)K + (unsigned)ak;       // 8B aligned
    unsigned aoff1 = aoff0 + ((unsigned)K << 4);                                 // +16 rows
    unsigned boff  = (((unsigned)(ak >> 1) * (unsigned)Nc + col0 + l15) << 1);  // 8B aligned
    const unsigned bstep = 4u * (unsigned)Nc;

    v8f acc0 = {}; v8f acc1 = {}; v8f acc2 = {}; v8f acc3 = {};
    v8f acc4 = {}; v8f acc5 = {}; v8f acc6 = {}; v8f acc7 = {};

    for (int k = 0; k < K; k += 4) {
        const v2f a0 = *(const v2f*)(A + aoff0);         // rows [row0, row0+16)
        const v2f a1 = *(const v2f*)(A + aoff1);         // rows [row0+16, row0+32)
        const v2f b0 = *(const v2f*)(Bp + boff);         // packed pair: rows kk,kk+1 at col
        const v2f b1 = *(const v2f*)(Bp + boff + 32);
        const v2f b2 = *(const v2f*)(Bp + boff + 64);
        const v2f b3 = *(const v2f*)(Bp + boff + 96);
        __builtin_prefetch(A + aoff0 + 64, 0, 3);        // near-cache stream prefetch
        __builtin_prefetch(A + aoff1 + 64, 0, 3);
        // (neg_a, A, neg_b, B, c_mod, C, reuse_a, reuse_b)
        acc0 = __builtin_amdgcn_wmma_f32_16x16x4_f32(false, a0, false, b0, (short)0, acc0, false, false);
        acc1 = __builtin_amdgcn_wmma_f32_16x16x4_f32(false, a0, false, b1, (short)0, acc1, false, false);
        acc2 = __builtin_amdgcn_wmma_f32_16x16x4_f32(false, a0, false, b2, (short)0, acc2, false, false);
        acc3 = __builtin_amdgcn_wmma_f32_16x16x4_f32(false, a0, false, b3, (short)0, acc3, false, false);
        acc4 = __builtin_amdgcn_wmma_f32_16x16x4_f32(false, a1, false, b0, (short)0, acc4, false, false);
        acc5 = __builtin_amdgcn_wmma_f32_16x16x4_f32(false, a1, false, b1, (short)0, acc5, false, false);
        acc6 = __builtin_amdgcn_wmma_f32_16x16x4_f32(false, a1, false, b2, (short)0, acc6, false, false);
        acc7 = __builtin_amdgcn_wmma_f32_16x16x4_f32(false, a1, false, b3, (short)0, acc7, false, false);
        aoff0 += 4u;
        aoff1 += 4u;
        boff  += bstep;
    }

    // D layout: VGPR r, lanes 0-15 -> (M=r, N=lane); lanes 16-31 -> (M=8+r, N=lane-16)
    float* crow0 = C + (long)(row0 + (half << 3)) * Nc + col0 + l15;
    float* crow1 = crow0 + (long)16 * Nc;
#pragma unroll
    for (int r = 0; r < 8; ++r) {
        crow0[0]  = acc0[r];
        crow0[16] = acc1[r];
        crow0[32] = acc2[r];
        crow0[48] = acc3[r];
        crow1[0]  = acc4[r];
        crow1[16] = acc5[r];
        crow1[32] = acc6[r];
        crow1[48] = acc7[r];
        crow0 += Nc;
        crow1 += Nc;
    }
}

// ---------------- aggregation ----------------
// agg[i, c] = h[i, c] * dinv[i]^2   (self-loop term; initializes agg). h has ld=ldh.
__global__ void self_init(const float* __restrict__ h, const float* __restrict__ dinv,
                          float* __restrict__ agg, int fo, int ldh) {
    const int c   = blockIdx.x * blockDim.x + threadIdx.x;
    const int row = blockIdx.y;
    if (c < fo) {
        const float dv = dinv[row];
        agg[(long)row * fo + c] = h[(long)row * ldh + c] * dv * dv;
    }
}

// One wave per edge: agg[dst, :] += h[src, :] * dinv[src]*dinv[dst]
__global__ void edge_scatter(const long long* __restrict__ src, const long long* __restrict__ dst,
                             const float* __restrict__ dinv, const float* __restrict__ h,
                             float* __restrict__ agg, int e, int fo, int ldh) {
    const int lane = threadIdx.x;                            // 32 lanes
    const int ei   = blockIdx.x * blockDim.y + threadIdx.y;
    if (ei >= e) return;                                     // wave-uniform
    const int s = (int)src[ei];                              // wave-uniform loads
    const int d = (int)dst[ei];
    const float nrm = dinv[s] * dinv[d];
    const float4* hs = (const float4*)(h + (long)s * ldh);   // ldh%32==0 -> 16B aligned
    float* ad = agg + (long)d * fo;
    const int n4 = fo >> 2;
    for (int j = lane; j < n4; j += 32) {
        const float4 v = hs[j];
        float* a = ad + (j << 2);
        atomicAdd(a + 0, v.x * nrm);
        atomicAdd(a + 1, v.y * nrm);
        atomicAdd(a + 2, v.z * nrm);
        atomicAdd(a + 3, v.w * nrm);
    }
}

__global__ void bias_relu(float* __restrict__ agg, const float* __restrict__ b, int fo) {
    const int c   = blockIdx.x * blockDim.x + threadIdx.x;
    const int row = blockIdx.y;
    if (c < fo) {
        const long i = (long)row * fo + c;
        const float v = agg[i] + b[c];
        agg[i] = v > 0.0f ? v : 0.0f;
    }
}

// ---------------- pooling + head ----------------
// post-ReLU values are >= 0, so float max == unsigned-bits max; pooled pre-filled with +0.0
__global__ void pool_max(const float* __restrict__ act, const long long* __restrict__ batch,
                         float* __restrict__ pooled) {
    const int c   = threadIdx.x;
    const int row = blockIdx.x;
    if (c < 100) {
        const int g = (int)batch[row];
        const unsigned bits = __float_as_uint(act[(long)row * 100 + c]);
        atomicMax((unsigned*)&pooled[g * 100 + c], bits);
    }
}

__global__ void head_kernel(const float* __restrict__ pooled, const float* __restrict__ w,
                            const float* __restrict__ b, float* __restrict__ out) {
    const int g = blockIdx.x;
    const int lane = threadIdx.x;       // 32 lanes
    float s = 0.0f;
    for (int c = lane; c < 100; c += 32) s += pooled[g * 100 + c] * w[c];
    for (int off = 16; off > 0; off >>= 1) s += __shfl_down(s, off, 32);
    if (lane == 0) out[g] = 1.0f / (1.0f + expf(-(s + b[0])));
}

// ---------------- launcher ----------------
extern "C" void kernel_launch(void* const* d_in, const int* in_sizes, int n_in,
                              void* d_out, int out_size, void* d_ws, size_t ws_size,
                              hipStream_t stream) {
    const float*     x     = (const float*)d_in[0];
    const long long* eidx  = (const long long*)d_in[1];
    const long long* batch = (const long long*)d_in[2];
    const float* W[5]  = {(const float*)d_in[3], (const float*)d_in[5], (const float*)d_in[7],
                          (const float*)d_in[9], (const float*)d_in[11]};
    const float* Bi[5] = {(const float*)d_in[4], (const float*)d_in[6], (const float*)d_in[8],
                          (const float*)d_in[10], (const float*)d_in[12]};
    const float* lin_w = (const float*)d_in[13];
    const float* lin_b = (const float*)d_in[14];
    float*       out   = (float*)d_out;

    const long long* src = eidx;              // edge_index[0, :]
    const long long* dst = eidx + N_EDGES;    // edge_index[1, :]

    const int dims[6] = {19, 500, 400, 300, 200, 100};
    const int fiP[5]  = {20, 500, 400, 300, 200};        // K padded to multiple of 4
    const int foP[5]  = {512, 448, 320, 256, 128};       // Nc padded to multiple of 64

    // workspace carving (all 256B aligned)
    char* ws = (char*)d_ws;
    auto carve = [&](size_t bytes) -> char* {
        char* p = ws; ws += (bytes + 255) / 256 * 256; return p;
    };
    float* dinv   = (float*)carve((size_t)N_NODES * 4);
    float* pooled = (float*)carve((size_t)N_GRAPH * 100 * 4);
    float* xp     = (float*)carve((size_t)N_NODES * 20 * 4);      // x padded to K=20
    float* Wp[5];
    for (int l = 0; l < 5; ++l) Wp[l] = (float*)carve((size_t)fiP[l] * foP[l] * 4);
    float* hbuf   = (float*)carve((size_t)N_NODES * 512 * 4);     // pre-agg GEMM out (ld=foP)
    float* aggA   = (float*)carve((size_t)N_NODES * 500 * 4);     // layer outs 0,2,4
    float* aggB   = (float*)carve((size_t)N_NODES * 400 * 4);     // layer outs 1,3

    // zero-padded copy of x; zero-padded packed copies of the weights
    {
        dim3 g(1, N_NODES);
        pad_copy<<<g, 32, 0, stream>>>(x, xp, N_NODES, 19, 20);
        for (int l = 0; l < 5; ++l) {
            dim3 gw((foP[l] + 255) / 256, fiP[l]);
            pad_pack_W<<<gw, 256, 0, stream>>>(W[l], Wp[l], dims[l], dims[l + 1], foP[l]);
        }
    }

    // degree -> dinv
    fill_f32<<<(N_NODES + 255) / 256, 256, 0, stream>>>(dinv, N_NODES, 0.0f);
    count_deg<<<(N_EDGES + 255) / 256, 256, 0, stream>>>(dst, dinv, N_EDGES);
    make_dinv<<<(N_NODES + 255) / 256, 256, 0, stream>>>(dinv, N_NODES);

    const float* cur = xp;
    for (int l = 0; l < 5; ++l) {
        const int fo = dims[l + 1];
        const int K  = fiP[l];        // A leading dim == K for xp and exact-stride agg
        const int Nc = foP[l];
        float* agg = (l & 1) ? aggB : aggA;

        const int mTiles = N_NODES / 32;                 // 625 wave tiles of 32 rows
        dim3 gb(32, 4);
        dim3 gg((mTiles + 3) / 4, Nc / 64);
        gemm_wmma_f32<<<gg, gb, 0, stream>>>(cur, Wp[l], hbuf, N_NODES, K, Nc);

        dim3 sb(256);
        dim3 sg((fo + 255) / 256, N_NODES);
        self_init<<<sg, sb, 0, stream>>>(hbuf, dinv, agg, fo, Nc);

        dim3 eb(32, 8);                                  // one wave per edge
        dim3 eg((N_EDGES + 7) / 8, 1);
        edge_scatter<<<eg, eb, 0, stream>>>(src, dst, dinv, hbuf, agg, N_EDGES, fo, Nc);

        bias_relu<<<sg, sb, 0, stream>>>(agg, Bi[l], fo);
        cur = agg;
    }

    fill_f32<<<(N_GRAPH * 100 + 255) / 256, 256, 0, stream>>>(pooled, N_GRAPH * 100, 0.0f);
    pool_max<<<N_NODES, 128, 0, stream>>>(cur, batch, pooled);
    head_kernel<<<N_GRAPH, 32, 0, stream>>>(pooled, lin_w, lin_b, out);
}